// Gemma4MTPMaskedEmbedder_59554016526576
// MI455X (gfx1250) — compile-verified
//
#include <hip/hip_runtime.h>

// ---------------- problem constants ----------------
#define NTOK   128
#define HID    1024
#define NCENT  512
#define TOPK   8
#define VOCAB  131072
#define VPC    256              // vocab ids per centroid
#define FILLV  (-3.402823466e38f)   // jnp.finfo(float32).min

typedef float v2f __attribute__((ext_vector_type(2)));
typedef float v8f __attribute__((ext_vector_type(8)));

// ---------------------------------------------------------------------------
// Kernel 1: fill output with -FLT_MAX using 128-bit stores (64 MB -> HBM BW)
// ---------------------------------------------------------------------------
__global__ void fill_out_kernel(float4* __restrict__ out, int n4) {
    const float4 v = make_float4(FILLV, FILLV, FILLV, FILLV);
    int i = blockIdx.x * blockDim.x + threadIdx.x;
    int stride = gridDim.x * blockDim.x;
    for (; i < n4; i += stride) out[i] = v;
}

// ---------------------------------------------------------------------------
// Kernel 2: zero the per-cluster counters
// ---------------------------------------------------------------------------
__global__ void zero_counts_kernel(int* __restrict__ counts) {
    int i = blockIdx.x * blockDim.x + threadIdx.x;
    if (i < NCENT) counts[i] = 0;
}

// ---------------------------------------------------------------------------
// Kernel 3: scores = H(128x1024) @ C^T -> (128x512), f32 WMMA 16x16x4
// grid (8, 32), one wave32 per 16x16 tile
// ---------------------------------------------------------------------------
__global__ __launch_bounds__(32)
void scores_wmma_kernel(const float* __restrict__ H,
                        const float* __restrict__ Cw,
                        float* __restrict__ S) {
    const int lane = threadIdx.x;          // 0..31
    const int mt = blockIdx.x;             // token tile   (8)
    const int nt = blockIdx.y;             // centroid tile(32)
    const int l16 = lane & 15;
    const int kb  = (lane >> 4) << 1;      // 0 or 2

    const float* arow = H  + (size_t)(mt * 16 + l16) * HID;  // A row = token
    const float* brow = Cw + (size_t)(nt * 16 + l16) * HID;  // B col = centroid

    v8f acc = {};
#pragma unroll 4
    for (int k0 = 0; k0 < HID; k0 += 4) {
        v2f a = *(const v2f*)(arow + k0 + kb);
        v2f b = *(const v2f*)(brow + k0 + kb);
        acc = __builtin_amdgcn_wmma_f32_16x16x4_f32(
            /*neg_a=*/false, a, /*neg_b=*/false, b,
            /*c_mod=*/(short)0, acc, /*reuse_a=*/false, /*reuse_b=*/false);
    }
    // D layout: VGPR i -> row (lane>=16?8:0)+i, col lane%16
    const int mbase = mt * 16 + ((lane >> 4) << 3);
    const int ncol  = nt * 16 + l16;
#pragma unroll
    for (int i = 0; i < 8; ++i)
        S[(size_t)(mbase + i) * NCENT + ncol] = acc[i];
}

// ---------------------------------------------------------------------------
// Kernel 4: per-token top-8 over 512 scores (1 wave32 per token) and build
// cluster -> token lists via atomics.
// ---------------------------------------------------------------------------
__global__ __launch_bounds__(32)
void topk_kernel(const float* __restrict__ S,
                 int* __restrict__ counts,
                 int* __restrict__ lists) {
    const int t = blockIdx.x;
    const int lane = threadIdx.x;
    float s[16];
#pragma unroll
    for (int i = 0; i < 16; ++i) s[i] = S[(size_t)t * NCENT + i * 32 + lane];

    for (int k = 0; k < TOPK; ++k) {
        float best = -3.4e38f;
        int bidx = 0x7FFFFFFF;
#pragma unroll
        for (int i = 0; i < 16; ++i) {
            int idx = i * 32 + lane;
            if (s[i] > best || (s[i] == best && idx < bidx)) { best = s[i]; bidx = idx; }
        }
        // wave32 argmax reduction
        for (int off = 16; off > 0; off >>= 1) {
            float ob = __shfl_xor(best, off, 32);
            int   oi = __shfl_xor(bidx, off, 32);
            if (ob > best || (ob == best && oi < bidx)) { best = ob; bidx = oi; }
        }
        // all lanes agree on bidx now; owner knocks it out
        if (lane == (bidx & 31)) s[bidx >> 5] = -3.4e38f;
        if (lane == 0) {
            int slot = atomicAdd(&counts[bidx], 1);
            lists[bidx * NTOK + slot] = t;
        }
    }
}

// ---------------------------------------------------------------------------
// Kernel 5: grouped logits GEMM. block = (ntile 0..15, cluster 0..511).
// B = 16 gathered embedding rows (read from HBM exactly once per block),
// A = tokens that selected this cluster (zero-padded to tiles of 16).
// D tile (tokens x 16 vocab cols) scattered into out[token][vocab_id].
// ---------------------------------------------------------------------------
__global__ __launch_bounds__(32)
void logits_wmma_kernel(const float* __restrict__ H,
                        const float* __restrict__ W,
                        const int* __restrict__ ord,
                        const int* __restrict__ counts,
                        const int* __restrict__ lists,
                        float* __restrict__ out) {
    const int c  = blockIdx.y;             // cluster
    const int nt = blockIdx.x;             // 16-row tile within cluster
    const int cnt = counts[c];             // uniform across wave
    if (cnt == 0) return;

    const int lane = threadIdx.x;
    const int l16  = lane & 15;
    const int kb   = (lane >> 4) << 1;     // 0 or 2

    const int vid = ord[c * VPC + nt * 16 + l16];         // this lane's vocab col
    const float* brow = W + (size_t)vid * HID;
    const int* lst = lists + c * NTOK;

    for (int m0 = 0; m0 < cnt; m0 += 16) {
        const int mi  = m0 + l16;                         // A row this lane feeds
        const int tok = (mi < cnt) ? lst[mi] : -1;
        const float* arow = H + (size_t)(tok < 0 ? 0 : tok) * HID;

        v8f acc = {};
#pragma unroll 4
        for (int k0 = 0; k0 < HID; k0 += 4) {
            v2f a = {0.0f, 0.0f};
            if (tok >= 0) a = *(const v2f*)(arow + k0 + kb);   // reconverges before WMMA
            v2f b = *(const v2f*)(brow + k0 + kb);
            acc = __builtin_amdgcn_wmma_f32_16x16x4_f32(
                false, a, false, b, (short)0, acc, false, false);
        }
        const int mbase = m0 + ((lane >> 4) << 3);
#pragma unroll
        for (int i = 0; i < 8; ++i) {
            const int m = mbase + i;
            if (m < cnt) {
                const int tk = lst[m];
                out[(size_t)tk * VOCAB + vid] = acc[i];
            }
        }
    }
}

// ---------------------------------------------------------------------------
extern "C" void kernel_launch(void* const* d_in, const int* in_sizes, int n_in,
                              void* d_out, int out_size, void* d_ws, size_t ws_size,
                              hipStream_t stream) {
    const float* H   = (const float*)d_in[0];   // (128,1024)
    const float* W   = (const float*)d_in[1];   // (131072,1024)
    const float* Cw  = (const float*)d_in[2];   // (512,1024)
    const int*   ord = (const int*)d_in[3];     // (131072,)
    float* out = (float*)d_out;                 // (128,131072)

    // workspace layout
    char* ws = (char*)d_ws;
    float* S      = (float*)(ws);                       // 128*512 f32   = 256 KB
    int*   counts = (int*)  (ws + 256 * 1024);          // 512 int       =   2 KB
    int*   lists  = (int*)  (ws + 260 * 1024);          // 512*128 int   = 256 KB

    // 1) fill output with -FLT_MAX (64 MB)
    const int n4 = (NTOK * VOCAB) / 4;
    fill_out_kernel<<<4096, 256, 0, stream>>>((float4*)out, n4);

    // 2) zero per-cluster counters
    zero_counts_kernel<<<2, 256, 0, stream>>>(counts);

    // 3) centroid scores via f32 WMMA
    scores_wmma_kernel<<<dim3(NTOK / 16, NCENT / 16), 32, 0, stream>>>(H, Cw, S);

    // 4) top-8 + cluster->token lists
    topk_kernel<<<NTOK, 32, 0, stream>>>(S, counts, lists);

    // 5) grouped logits GEMM + scatter
    logits_wmma_kernel<<<dim3(VPC / 16, NCENT), 32, 0, stream>>>(
        H, W, ord, counts, lists, out);
}